// ChamferLoss_75290776699469
// MI455X (gfx1250) — compile-verified
//
#include <hip/hip_runtime.h>
#include <hip/hip_bf16.h>

typedef float v2f __attribute__((ext_vector_type(2)));
typedef float v8f __attribute__((ext_vector_type(8)));

static constexpr int BB = 16;
static constexpr int MM = 4096;
static constexpr int NN = 4096;
static constexpr unsigned INF_BITS = 0x7F800000u;

// ---------------------------------------------------------------------------
// Kernel 1: seed the backward-min region of d_out with +inf bit patterns.
// ---------------------------------------------------------------------------
__global__ void chamfer_init_kernel(unsigned* __restrict__ bwd_bits, int n) {
    int idx = blockIdx.x * blockDim.x + threadIdx.x;
    if (idx < n) bwd_bits[idx] = INF_BITS;
}

// ---------------------------------------------------------------------------
// Kernel 2: main WMMA kernel.
//   d2[m,n] = |t_m|^2 + |a_n|^2 - 2*(t_m . a_n)
// Cross term via V_WMMA_F32_16X16X4_F32 (K padded 2->4 with zeros).
// Block = 8 waves; wave w owns M-tile [m0, m0+16), sweeps all N.
// Software-pipelined: 4 B-fragment loads for iteration i+1 are issued before
// the 4 WMMA+epilogue blocks of iteration i, hiding global load latency.
// Forward min: register accumulators. Backward min: all 32 lanes ds_min_u32
// into slot n0+(lane&15) (lo half = rows 0-7 min, hi half = rows 8-15 min).
// ---------------------------------------------------------------------------
__global__ void __launch_bounds__(256)
chamfer_wmma_kernel(const float* __restrict__ target,
                    const float* __restrict__ actual,
                    float* __restrict__ fwd_sq,        // [B*M] squared dists
                    unsigned* __restrict__ bwd_bits) { // [B*N] float bits
    __shared__ unsigned ldsb[NN]; // 16 KB of the 320 KB WGP LDS

    const int tid  = threadIdx.x;
    const int wave = tid >> 5;
    const int lane = tid & 31;
    const bool lo  = lane < 16;
    const int col  = lane & 15;

    const int blocks_per_b = MM / 128;                 // 32
    const int b  = blockIdx.x / blocks_per_b;
    const int m0 = (blockIdx.x % blocks_per_b) * 128 + wave * 16;

    // init LDS backward accumulator
    for (int i = tid; i < NN; i += 256) ldsb[i] = INF_BITS;
    __syncthreads();

    const float2* tp2 = reinterpret_cast<const float2*>(target) + (size_t)b * MM;
    const float2* ap2 = reinterpret_cast<const float2*>(actual) + (size_t)b * NN;

    // --- A fragment (16x4 f32, K=2 real, K=2..3 zero): lanes 0-15 = rows
    const float2 t = tp2[m0 + col];
    v2f afrag;
    afrag.x = lo ? t.x : 0.0f;
    afrag.y = lo ? t.y : 0.0f;

    // row norms broadcast to match C layout: VGPR r, lane<16 -> row r, lane>=16 -> row r+8
    const float tnorm = fmaf(t.x, t.x, t.y * t.y);
    float tn[8];
#pragma unroll
    for (int r = 0; r < 8; ++r)
        tn[r] = __shfl(tnorm, r + ((lane & 16) >> 1), 32);

    float fwd_acc[8];
#pragma unroll
    for (int r = 0; r < 8; ++r) fwd_acc[r] = __builtin_inff();

    // one 16x16 tile: WMMA dot-product tile + distance epilogue + backward atomic
    auto tile = [&](float2 a, int n0) {
        v2f bfrag;
        bfrag.x = lo ? a.x : 0.0f;
        bfrag.y = lo ? a.y : 0.0f;
        const float anorm = fmaf(a.x, a.x, a.y * a.y); // col norm (col = lane&15)

        v8f c = {};
        c = __builtin_amdgcn_wmma_f32_16x16x4_f32(
                /*neg_a=*/false, afrag, /*neg_b=*/false, bfrag,
                /*c_mod=*/(short)0, c, /*reuse_a=*/false, /*reuse_b=*/false);

        float colmin = __builtin_inff();
#pragma unroll
        for (int r = 0; r < 8; ++r) {
            float d2 = fmaf(c[r], -2.0f, tn[r] + anorm);
            d2 = fmaxf(d2, 0.0f);                  // guard fp cancellation
            fwd_acc[r] = fminf(fwd_acc[r], d2);
            colmin     = fminf(colmin, d2);
        }
        // lo half = min over rows 0-7, hi half = min over rows 8-15; LDS atomic
        // fuses the two lanes hitting the same slot -> full column min.
        atomicMin(&ldsb[n0 + col], __float_as_uint(colmin)); // ds_min_u32
    };

    // --- software-pipelined sweep over N: prefetch next 4 tiles, compute 4
    float2 n0v = ap2[col];
    float2 n1v = ap2[16 + col];
    float2 n2v = ap2[32 + col];
    float2 n3v = ap2[48 + col];

    for (int n0 = 0; n0 < NN; n0 += 64) {
        const float2 c0 = n0v, c1 = n1v, c2 = n2v, c3 = n3v;
        const int np = (n0 + 64) & (NN - 1); // wraps to 0 on last iter (harmless reload)
        n0v = ap2[np + col];
        n1v = ap2[np + 16 + col];
        n2v = ap2[np + 32 + col];
        n3v = ap2[np + 48 + col];

        tile(c0, n0);
        tile(c1, n0 + 16);
        tile(c2, n0 + 32);
        tile(c3, n0 + 48);
    }

    // --- forward reduction: min over the 16 lanes of each half (cols), per row
#pragma unroll
    for (int r = 0; r < 8; ++r) {
        float v = fwd_acc[r];
        v = fminf(v, __shfl_xor(v, 1, 32));
        v = fminf(v, __shfl_xor(v, 2, 32));
        v = fminf(v, __shfl_xor(v, 4, 32));
        v = fminf(v, __shfl_xor(v, 8, 32));
        if (lane == 0)  fwd_sq[(size_t)b * MM + m0 + r]     = v; // row r   (lo half)
        if (lane == 16) fwd_sq[(size_t)b * MM + m0 + 8 + r] = v; // row r+8 (hi half)
    }

    // --- flush backward partials (8x fewer global atomics than per-wave)
    __syncthreads();
    for (int i = tid; i < NN; i += 256)
        atomicMin(&bwd_bits[(size_t)b * NN + i], ldsb[i]); // global_atomic_min_u32
}

// ---------------------------------------------------------------------------
// Kernel 3: finalize in place. Both regions hold nonneg f32 bit patterns of
// squared distances -> reinterpret + sqrt.
// ---------------------------------------------------------------------------
__global__ void chamfer_finalize_kernel(unsigned* __restrict__ io, int n) {
    int idx = blockIdx.x * blockDim.x + threadIdx.x;
    if (idx < n) {
        float v = __uint_as_float(io[idx]);
        reinterpret_cast<float*>(io)[idx] = sqrtf(v);
    }
}

extern "C" void kernel_launch(void* const* d_in, const int* in_sizes, int n_in,
                              void* d_out, int out_size, void* d_ws, size_t ws_size,
                              hipStream_t stream) {
    const float* target = (const float*)d_in[0]; // [B, M, 2]
    const float* actual = (const float*)d_in[1]; // [B, N, 2]

    float*    fwd_sq   = (float*)d_out;                  // first B*M floats
    unsigned* bwd_bits = (unsigned*)d_out + BB * MM;     // next  B*N uints (bit-min)

    // 1) seed backward region with +inf
    {
        int n = BB * NN;
        chamfer_init_kernel<<<(n + 255) / 256, 256, 0, stream>>>(bwd_bits, n);
    }
    // 2) main WMMA pass: one block per (b, 128-row M group)
    {
        int grid = BB * (MM / 128); // 512 blocks x 8 waves
        chamfer_wmma_kernel<<<grid, 256, 0, stream>>>(target, actual, fwd_sq, bwd_bits);
    }
    // 3) sqrt in place over both outputs
    {
        int n = BB * MM + BB * NN;
        chamfer_finalize_kernel<<<(n + 255) / 256, 256, 0, stream>>>((unsigned*)d_out, n);
    }
}